// CAMLoss_11390253269662
// MI455X (gfx1250) — compile-verified
//
#include <hip/hip_runtime.h>
#include <hip/hip_bf16.h>
#include <math.h>

typedef float v2f __attribute__((ext_vector_type(2)));
typedef float v8f __attribute__((ext_vector_type(8)));

#define BZ    64
#define NC    2048
#define H     14
#define W     14
#define HW    196
#define NCLS  1000
#define PD_EPS 1e-6f
#define MARGIN 70.0f
#define THR    125.0f

// ---- workspace layout (floats) ----
#define OFF_CAMRAW 0                        // [64][3][256] raw cam accumulators (first 196 used)
#define OFF_CAMN   (OFF_CAMRAW + BZ*3*256)  // [64][3][256] normalized cam, zero-padded 16x16
#define OFF_DELTA  (OFF_CAMN   + BZ*3*256)  // [64][2][256] delta01/delta02, zero-padded 16x16
#define OFF_ED1    (OFF_DELTA  + BZ*2*256)  // [64]
#define OFF_S01    (OFF_ED1 + BZ)           // [64]
#define OFF_S02    (OFF_S01 + BZ)           // [64]
#define OFF_LOSS1  (OFF_S02 + BZ)           // [1]
#define WS_FLOATS  (OFF_LOSS1 + 1)          // ~513 KB total

__global__ void k_zero(float* ws) {
  int i = blockIdx.x * blockDim.x + threadIdx.x;
  if (i < WS_FLOATS) ws[i] = 0.0f;
}

// ---- cross entropy: loss1 = -mean(log_softmax(pred)[b, cla[b]]) ----
__global__ void k_ce(const float* __restrict__ pred, const int* __restrict__ cla, float* ws) {
  int b = blockIdx.x, t = threadIdx.x;
  __shared__ float sh[256];
  const float* row = pred + (size_t)b * NCLS;
  float mx = -3.4e38f;
  for (int i = t; i < NCLS; i += 256) mx = fmaxf(mx, row[i]);
  sh[t] = mx; __syncthreads();
  for (int s = 128; s > 0; s >>= 1) { if (t < s) sh[t] = fmaxf(sh[t], sh[t + s]); __syncthreads(); }
  mx = sh[0]; __syncthreads();
  float sum = 0.0f;
  for (int i = t; i < NCLS; i += 256) sum += __expf(row[i] - mx);
  sh[t] = sum; __syncthreads();
  for (int s = 128; s > 0; s >>= 1) { if (t < s) sh[t] += sh[t + s]; __syncthreads(); }
  if (t == 0) {
    float logp = row[cla[b]] - mx - logf(sh[0]);
    atomicAdd(&ws[OFF_LOSS1], -logp / (float)BZ);
  }
}

// ---- CAM accumulation: cam_k[b,p] = sum_c w[idx[b,k],c] * F[b,c,p] ----
// grid (64, 16); each block handles 128 channels; pixel-per-thread (coalesced along p)
__global__ void k_cam_accum(const float* __restrict__ F, const float* __restrict__ wsm,
                            const int* __restrict__ idx, float* ws) {
  int b = blockIdx.x, cg = blockIdx.y, p = threadIdx.x;
  if (p >= HW) return;
  int w0 = idx[b * 3 + 0], w1 = idx[b * 3 + 1], w2 = idx[b * 3 + 2];
  const float* fb = F + (size_t)b * NC * HW + p;
  float a0 = 0.f, a1 = 0.f, a2 = 0.f;
  int c0 = cg * 128;
  for (int c = c0; c < c0 + 128; ++c) {
    float f = fb[(size_t)c * HW];                 // per-thread coalesced
    a0 = fmaf(wsm[(size_t)w0 * NC + c], f, a0);   // uniform -> scalar loads
    a1 = fmaf(wsm[(size_t)w1 * NC + c], f, a1);
    a2 = fmaf(wsm[(size_t)w2 * NC + c], f, a2);
  }
  atomicAdd(&ws[OFF_CAMRAW + (b * 3 + 0) * 256 + p], a0);
  atomicAdd(&ws[OFF_CAMRAW + (b * 3 + 1) * 256 + p], a1);
  atomicAdd(&ws[OFF_CAMRAW + (b * 3 + 2) * 256 + p], a2);
}

// ---- normalize: cam = (cam - min); cam = cam / max * 255; store padded 16x16 ----
__global__ void k_cam_norm(float* ws) {
  int b = blockIdx.x, k = blockIdx.y, t = threadIdx.x;
  __shared__ float sh[256];
  float v = (t < HW) ? ws[OFF_CAMRAW + (b * 3 + k) * 256 + t] : 3.4e38f;
  sh[t] = v; __syncthreads();
  for (int s = 128; s > 0; s >>= 1) { if (t < s) sh[t] = fminf(sh[t], sh[t + s]); __syncthreads(); }
  float mn = sh[0]; __syncthreads();
  float u = v - mn;
  sh[t] = (t < HW) ? u : -3.4e38f; __syncthreads();
  for (int s = 128; s > 0; s >>= 1) { if (t < s) sh[t] = fmaxf(sh[t], sh[t + s]); __syncthreads(); }
  float mx = sh[0];
  if (t < HW) {
    int r = t / W, c = t % W;
    ws[OFF_CAMN + (b * 3 + k) * 256 + r * 16 + c] = u / mx * 255.0f;
  }
}

// ---- delta01/02 (padded) + ed1[b] = mean_rows ||dst - seg + eps||_2 ----
__global__ void k_delta_ed1(const float* __restrict__ seg, float* ws) {
  int b = blockIdx.x, t = threadIdx.x;
  const float* cam0 = ws + OFF_CAMN + (b * 3 + 0) * 256;
  const float* cam1 = ws + OFF_CAMN + (b * 3 + 1) * 256;
  const float* cam2 = ws + OFF_CAMN + (b * 3 + 2) * 256;
  ws[OFF_DELTA + (b * 2 + 0) * 256 + t] = cam0[t] - cam1[t];  // padding stays 0
  ws[OFF_DELTA + (b * 2 + 1) * 256 + t] = cam0[t] - cam2[t];
  __shared__ float sh[256];
  float term = 0.0f;
  if (t < HW) {
    int r = t / W, c = t % W;
    float dst = (cam0[r * 16 + c] > THR) ? 1.0f : 0.0f;
    float d = dst - seg[(size_t)b * HW + t] + PD_EPS;
    term = d * d;
  }
  sh[t] = term; __syncthreads();
  float rs = 0.0f;
  if (t < H) { for (int c = 0; c < W; ++c) rs += sh[t * W + c]; }
  __syncthreads();
  if (t < H) sh[t] = sqrtf(rs);
  __syncthreads();
  if (t == 0) {
    float s = 0.0f;
    for (int i = 0; i < H; ++i) s += sh[i];
    ws[OFF_ED1 + b] = s / (float)H;
  }
}

// ---- WMMA kernel: s01[b] = sum_c sum_{i<14,j<14} ((D01 @ F_c)[i,j] + eps)^2 (ditto s02) ----
// grid (64, 32), 256 thr = 8 waves; 256 waves per batch; 8 channels per wave.
// Product P = Delta(16x16 padded) @ F_c(16x16 padded) via 4 chained V_WMMA_F32_16X16X4_F32.
// All padding handled by branchless clamped loads + 0/1 mask multiplies (EXEC stays all-1s).
__global__ void __launch_bounds__(256) k_fea_dist(const float* __restrict__ F, float* ws) {
  int b = blockIdx.x;
  int wave = blockIdx.y * 8 + (threadIdx.x >> 5);
  int lane = threadIdx.x & 31;
  int hi = lane >> 4;        // half-wave select
  int n  = lane & 15;        // A: M index; B/C: N index
  const float* d01 = ws + OFF_DELTA + (b * 2 + 0) * 256;
  const float* d02 = ws + OFF_DELTA + (b * 2 + 1) * 256;

  // A fragments (16x4 f32, 2 VGPRs): lanes0-15 -> K={4t,4t+1}, lanes16-31 -> K={4t+2,4t+3}
  v2f a01[4], a02[4];
#pragma unroll
  for (int t = 0; t < 4; ++t) {
    int kk = 4 * t + hi * 2;
    a01[t].x = d01[n * 16 + kk];     a01[t].y = d01[n * 16 + kk + 1];
    a02[t].x = d02[n * 16 + kk];     a02[t].y = d02[n * 16 + kk + 1];
  }

  // loop-invariant per-lane B offsets (clamped into the 196-float tile) and 0/1 masks
  int   offx[4], offy[4];
  float mvx[4],  mvy[4];
#pragma unroll
  for (int t = 0; t < 4; ++t) {
    int k0 = 4 * t + hi * 2;
    bool vx = (k0     < H) && (n < W);
    bool vy = (k0 + 1 < H) && (n < W);
    offx[t] = vx ? (k0 * W + n)       : 0;   // clamped -> always a legal in-tile load
    offy[t] = vy ? ((k0 + 1) * W + n) : 0;
    mvx[t]  = vx ? 1.0f : 0.0f;
    mvy[t]  = vy ? 1.0f : 0.0f;
  }

  float s01 = 0.0f, s02 = 0.0f;
  const float* fb = F + (size_t)b * NC * HW;
#pragma unroll 2
  for (int i = 0; i < 8; ++i) {
    int c = wave + 256 * i;
    const float* fc = fb + (size_t)c * HW;
    v8f p01 = {}; v8f p02 = {};
    v2f bf[4];
#pragma unroll
    for (int t = 0; t < 4; ++t) {            // issue all loads first (branchless)
      bf[t].x = fc[offx[t]] * mvx[t];
      bf[t].y = fc[offy[t]] * mvy[t];
    }
#pragma unroll
    for (int t = 0; t < 4; ++t) {
      p01 = __builtin_amdgcn_wmma_f32_16x16x4_f32(false, a01[t], false, bf[t],
                                                  (short)0, p01, false, false);
      p02 = __builtin_amdgcn_wmma_f32_16x16x4_f32(false, a02[t], false, bf[t],
                                                  (short)0, p02, false, false);
    }
    // padded rows (M>=14) / cols (N>=14) contribute 0 via mask factor g
#pragma unroll
    for (int e = 0; e < 8; ++e) {
      float g = (((e + hi * 8) < H) && (n < W)) ? 1.0f : 0.0f;
      float x = (p01[e] + PD_EPS) * g;  s01 = fmaf(x, x, s01);
      float y = (p02[e] + PD_EPS) * g;  s02 = fmaf(y, y, s02);
    }
  }

  __shared__ float r01[256], r02[256];
  int t = threadIdx.x;
  r01[t] = s01; r02[t] = s02; __syncthreads();
  for (int s = 128; s > 0; s >>= 1) {
    if (t < s) { r01[t] += r01[t + s]; r02[t] += r02[t + s]; }
    __syncthreads();
  }
  if (t == 0) {
    atomicAdd(&ws[OFF_S01 + b], r01[0]);
    atomicAdd(&ws[OFF_S02 + b], r02[0]);
  }
}

// ---- final: mean(ed1 + clip(70 - sqrt(s01)/nc - sqrt(s02)/nc, 0)) + loss1 ----
__global__ void k_final(const float* __restrict__ ws, float* __restrict__ out) {
  int t = threadIdx.x;
  __shared__ float sh[64];
  float term = 0.0f;
  if (t < BZ) {
    float d01 = sqrtf(ws[OFF_S01 + t]) / (float)NC;
    float d02 = sqrtf(ws[OFF_S02 + t]) / (float)NC;
    term = ws[OFF_ED1 + t] + fmaxf(MARGIN - d01 - d02, 0.0f);
  }
  sh[t] = term; __syncthreads();
  for (int s = 32; s > 0; s >>= 1) { if (t < s) sh[t] += sh[t + s]; __syncthreads(); }
  if (t == 0) out[0] = sh[0] / (float)BZ + ws[OFF_LOSS1];
}

extern "C" void kernel_launch(void* const* d_in, const int* in_sizes, int n_in,
                              void* d_out, int out_size, void* d_ws, size_t ws_size,
                              hipStream_t stream) {
  (void)in_sizes; (void)n_in; (void)out_size; (void)ws_size;
  const float* pred = (const float*)d_in[0];
  const int*   cla  = (const int*)  d_in[1];
  const float* seg  = (const float*)d_in[2];
  const float* F    = (const float*)d_in[3];
  const float* wsm  = (const float*)d_in[4];
  const int*   idx  = (const int*)  d_in[5];
  float* ws  = (float*)d_ws;
  float* out = (float*)d_out;

  k_zero     <<<(WS_FLOATS + 255) / 256, 256, 0, stream>>>(ws);
  k_ce       <<<BZ, 256, 0, stream>>>(pred, cla, ws);
  k_cam_accum<<<dim3(BZ, 16), 256, 0, stream>>>(F, wsm, idx, ws);
  k_cam_norm <<<dim3(BZ, 3), 256, 0, stream>>>(ws);
  k_delta_ed1<<<BZ, 256, 0, stream>>>(seg, ws);
  k_fea_dist <<<dim3(BZ, 32), 256, 0, stream>>>(F, ws);
  k_final    <<<1, 64, 0, stream>>>(ws, out);
}